// Linear_mask_18408229831014
// MI455X (gfx1250) — compile-verified
//
#include <hip/hip_runtime.h>

// Problem constants (match the reference).
#define Bb 256
#define Nn 1024
#define Ll 128
#define Mm 512   // Mm is a power of two: e/Mm == e>>9

typedef float v2f __attribute__((ext_vector_type(2)));
typedef float v8f __attribute__((ext_vector_type(8)));

// ---------------------------------------------------------------------------
// Pass 1a: zero per-row "masked" flags (one uint per row, B*N rows).
// ---------------------------------------------------------------------------
__global__ void zero_flags_kernel(unsigned* __restrict__ flags) {
    const int i = blockIdx.x * blockDim.x + threadIdx.x;
    flags[i] = 0u;
}

// ---------------------------------------------------------------------------
// Pass 1b: scatter flags[b*N + idx] = 1 for every (b, m) entry.
// Duplicates race writing the identical value 1 -> benign.
// ---------------------------------------------------------------------------
__global__ void scatter_flags_kernel(const int* __restrict__ idx,
                                     unsigned* __restrict__ flags) {
    const long long e = (long long)blockIdx.x * blockDim.x + threadIdx.x; // [0, B*M)
    const int b = (int)(e >> 9);                 // e / M
    const int row = idx[e];
    flags[(long long)b * Nn + row] = 1u;
}

// ---------------------------------------------------------------------------
// Pass 2: stream copy, one wave32 per 128-float row. Lane l moves float4 #l
// (32 lanes * 16B = 512B = one row, exact b128 load/store). Masked rows are
// skipped entirely (wave-uniform branch) -> no read, no write for them.
// ---------------------------------------------------------------------------
__global__ void copy_rows_kernel(const float4* __restrict__ src,
                                 float4* __restrict__ dst,
                                 const unsigned* __restrict__ flags,
                                 int useFlags) {
    const int lane = threadIdx.x & 31;
    const long long row = (long long)blockIdx.x * (blockDim.x >> 5) + (threadIdx.x >> 5);
    if (useFlags && flags[row]) return;          // row will be produced by WMMA pass
    const long long q = row * 32 + lane;         // float4 index
    dst[q] = src[q];
}

// ---------------------------------------------------------------------------
// Pass 3: generate masked rows with V_WMMA_F32_16X16X4_F32.
//   D[M,N] = start_M * (1 - t_N) + end_M * t_N
// A (16x4 f32): lanes 0-15 hold (K0,K1) = (start, end); lanes 16-31 hold
//               (K2,K3) = (0,0).
// B (4x16 f32): column n lives in lane n: lanes 0-15 = (1-t, t); lanes 16-31
//               (K2,K3 contribution) = (0,0).
// One wave handles 16 flat entries x 128 columns = 8 WMMAs.
// D layout: vgpr i, lane j -> entry (i + 8*(j>=16)), column (j & 15).
// ---------------------------------------------------------------------------
__global__ void lerp_wmma_kernel(const float* __restrict__ patches,
                                 const int* __restrict__ idx,
                                 float* __restrict__ out) {
    const int lane = threadIdx.x & 31;
    const int half = lane >> 4;      // 0: lanes 0-15, 1: lanes 16-31
    const int l15  = lane & 15;
    const long long wave  = (long long)blockIdx.x * (blockDim.x >> 5) + (threadIdx.x >> 5);
    const long long ebase = wave * 16;           // 16 entries per wave, total B*M

    // --- A operand: (start, end) for entry ebase+l15 in lanes 0-15 ---------
    v2f a;
    if (half == 0) {
        const long long e = ebase + l15;
        const int b = (int)(e >> 9);
        const int row = idx[e];
        const float* rp = patches + ((long long)b * Nn + row) * Ll;
        a.x = rp[0];
        a.y = rp[Ll - 1];
    } else {
        a.x = 0.0f;
        a.y = 0.0f;
    }

    // --- output row pointers: vgpr i of this lane belongs to entry
    //     ebase + i + 8*half -----------------------------------------------
    float* rowptr[8];
#pragma unroll
    for (int i = 0; i < 8; ++i) {
        const long long e = ebase + i + half * 8;
        const int b = (int)(e >> 9);
        const int row = idx[e];
        rowptr[i] = out + ((long long)b * Nn + row) * Ll;
    }

    // --- 8 column tiles of 16 -> full wave WMMA each -----------------------
#pragma unroll
    for (int tile = 0; tile < 8; ++tile) {
        v2f bv;
        if (half == 0) {
            const float t = (float)(tile * 16 + l15) * (1.0f / (float)(Ll - 1));
            bv.x = 1.0f - t;
            bv.y = t;
        } else {
            bv.x = 0.0f;
            bv.y = 0.0f;
        }
        v8f c = {};
        // 8 args: (neg_a, A, neg_b, B, c_mod, C, reuse_a, reuse_b)
        v8f d = __builtin_amdgcn_wmma_f32_16x16x4_f32(
            false, a, false, bv, (short)0, c, false, false);
#pragma unroll
        for (int i = 0; i < 8; ++i) {
            rowptr[i][tile * 16 + l15] = d[i];   // half-wave writes 64B contiguous
        }
    }
}

// ---------------------------------------------------------------------------
extern "C" void kernel_launch(void* const* d_in, const int* in_sizes, int n_in,
                              void* d_out, int out_size, void* d_ws, size_t ws_size,
                              hipStream_t stream) {
    const float* patches = (const float*)d_in[0];
    const int*   idx     = (const int*)d_in[1];
    float*       out     = (float*)d_out;

    const long long rows    = (long long)Bb * Nn;   // 262144
    const long long entries = (long long)Bb * Mm;   // 131072

    unsigned* flags = (unsigned*)d_ws;
    const int useFlags = (d_ws != nullptr &&
                          ws_size >= (size_t)rows * sizeof(unsigned)) ? 1 : 0;

    if (useFlags) {
        zero_flags_kernel<<<(int)(rows / 256), 256, 0, stream>>>(flags);
        scatter_flags_kernel<<<(int)(entries / 256), 256, 0, stream>>>(idx, flags);
    }

    // 256 threads = 8 waves = 8 rows per block.
    copy_rows_kernel<<<(int)(rows / 8), 256, 0, stream>>>(
        (const float4*)patches, (float4*)out, flags, useFlags);

    // 16 entries per wave, 8 waves per block -> 128 entries per block.
    lerp_wmma_kernel<<<(int)(entries / 128), 256, 0, stream>>>(patches, idx, out);
}